// iFFTlayer_25950192403256
// MI455X (gfx1250) — compile-verified
//
#include <hip/hip_runtime.h>
#include <hip/hip_bf16.h>

typedef __attribute__((ext_vector_type(2))) float v2f;
typedef __attribute__((ext_vector_type(8))) float v8f;

#define IN_FEAT   32768
#define NUMP      750
#define NP_PAD    752      // padded to 47 tiles of 16
#define NTILES    47
#define KCHUNK    512
#define KCHUNKS   (IN_FEAT / KCHUNK)   // 64

// ---------------------------------------------------------------------------
// Kernel 0: zero the split-K accumulator (must run every call: atomics below)
// ---------------------------------------------------------------------------
__global__ void zero_ws_kernel(float* __restrict__ p, int n) {
    int i = blockIdx.x * blockDim.x + threadIdx.x;
    if (i < n) p[i] = 0.0f;
}

// ---------------------------------------------------------------------------
// Kernel 1: x = image @ W via V_WMMA_F32_16X16X4_F32, split-K with f32 atomics
//   grid = (NTILES, KCHUNKS), block = 32 (one wave; EXEC all-ones for WMMA)
//   A (image) 32x32768 row-major, B (W) 32768x750 row-major.
//   fp32 A 16x4 layout: lanes 0-15 = M rows, VGPR0={K0|K2}, VGPR1={K1|K3}
//   fp32 B 4x16 layout: lanes 0-15 = N cols, VGPR0={K0|K2}, VGPR1={K1|K3}
//   fp32 C/D 16x16:     lane L, vgpr r -> (M = r + 8*(L/16), N = L%16)
// ---------------------------------------------------------------------------
__global__ void gemm_wmma_kernel(const float* __restrict__ img,
                                 const float* __restrict__ W,
                                 float* __restrict__ xbuf) {
    const int nt   = blockIdx.x;        // output column tile
    const int kc   = blockIdx.y;        // K chunk
    const int lane = threadIdx.x;       // 0..31
    const int half = lane >> 4;         // 0 or 1
    const int l16  = lane & 15;

    const int n0 = nt * 16;
    int colw = n0 + l16;                        // 0..751 (store column, padded)
    int colr = colw > (NUMP - 1) ? (NUMP - 1) : colw;  // clamp W reads in-bounds
    const int k0 = kc * KCHUNK;

    // A fragment pointers: row l16 (tile m0=0) and row 16+l16 (tile m0=16),
    // columns k0 + 2*half .. +1  (8-byte aligned -> b64 loads)
    const float* pa0 = img + (size_t)l16        * IN_FEAT + k0 + 2 * half;
    const float* pa1 = img + (size_t)(16 + l16) * IN_FEAT + k0 + 2 * half;
    // B fragment pointer: rows k0 + 2*half, k0 + 2*half + 1, column colr
    const float* pb  = W + (size_t)(k0 + 2 * half) * NUMP + colr;

    v8f acc0 = {};
    v8f acc1 = {};

    #pragma unroll 4
    for (int it = 0; it < KCHUNK / 4; ++it) {
        v2f a0 = *(const v2f*)pa0;
        v2f a1 = *(const v2f*)pa1;
        v2f bf;
        bf.x = pb[0];
        bf.y = pb[NUMP];
        acc0 = __builtin_amdgcn_wmma_f32_16x16x4_f32(
            false, a0, false, bf, (short)0, acc0, false, false);
        acc1 = __builtin_amdgcn_wmma_f32_16x16x4_f32(
            false, a1, false, bf, (short)0, acc1, false, false);
        pa0 += 4;
        pa1 += 4;
        pb  += 4 * NUMP;
    }

    // split-K reduction into xbuf (32 x NP_PAD)
    float* o0 = xbuf + (size_t)(half * 8)      * NP_PAD + colw;
    float* o1 = xbuf + (size_t)(16 + half * 8) * NP_PAD + colw;
    #pragma unroll
    for (int r = 0; r < 8; ++r) {
        atomicAdd(o0 + (size_t)r * NP_PAD, acc0[r]);
        atomicAdd(o1 + (size_t)r * NP_PAD, acc1[r]);
    }
}

// ---------------------------------------------------------------------------
// Kernel 2: sparse-FFT evaluated as separable 5-point DFTs, fused in LDS.
//   grid = (u=32, b=32), block = 256.
//   out[b,u,v,w,f] = Re+Im( sum_k c[b,f,k] * e^{-2pi i (u kx + v ky + w kz)/64} )
// ---------------------------------------------------------------------------
__global__ void ifft_eval_kernel(const float* __restrict__ xbuf,
                                 const float* __restrict__ bias,
                                 float* __restrict__ out) {
    __shared__ float c[NUMP];
    __shared__ float T1re[150], T1im[150];   // [f*25 + ky*5 + kz]
    __shared__ float T2re[960], T2im[960];   // [(f*5 + kz)*32 + v]
    __shared__ float Fp[160], Fm[160];       // [w*5 + kz]

    const int u = blockIdx.x;
    const int b = blockIdx.y;
    const int t = threadIdx.x;
    const float W64 = 6.28318530717958647692f / 64.0f;

    // coeffs = x[b,:] + bias
    for (int i = t; i < NUMP; i += 256)
        c[i] = xbuf[(size_t)b * NP_PAD + i] + bias[i];

    // kz-axis output factors: theta = 2pi*w*(kz-2)/64
    if (t < 160) {
        int w  = t / 5;
        int kz = t % 5;
        float s, co;
        __sincosf(W64 * (float)(w * (kz - 2)), &s, &co);
        Fp[t] = co - s;   // multiplies Re
        Fm[t] = co + s;   // multiplies Im
    }
    __syncthreads();

    // T1[f,ky,kz] = sum_kx c[f,kx,ky,kz] * e^{-i theta_ux}
    if (t < 150) {
        int f  = t / 25;
        int r  = t % 25;
        int ky = r / 5;
        int kz = r % 5;
        float sre = 0.0f, sim = 0.0f;
        #pragma unroll
        for (int a = 0; a < 5; ++a) {
            float s, co;
            __sincosf(W64 * (float)(u * (a - 2)), &s, &co);
            float cv = c[f * 125 + a * 25 + ky * 5 + kz];
            sre += cv * co;
            sim -= cv * s;
        }
        T1re[t] = sre;
        T1im[t] = sim;
    }
    __syncthreads();

    // T2[f,kz,v] = sum_ky T1[f,ky,kz] * e^{-i theta_vy}
    for (int i = t; i < 960; i += 256) {
        int f  = i / 160;
        int r  = i % 160;
        int kz = r / 32;
        int v  = r % 32;
        float sre = 0.0f, sim = 0.0f;
        #pragma unroll
        for (int ky = 0; ky < 5; ++ky) {
            float s, co;
            __sincosf(W64 * (float)(v * (ky - 2)), &s, &co);
            float ar = T1re[(f * 5 + ky) * 5 + kz];
            float ai = T1im[(f * 5 + ky) * 5 + kz];
            sre += ar * co + ai * s;     // (a+bi)(cos - i sin): Re
            sim += ai * co - ar * s;     //                      Im
        }
        T2re[i] = sre;
        T2im[i] = sim;
    }
    __syncthreads();

    // out slab for (b,u): 32*32*6 floats, coalesced stores
    float* obase = out + ((size_t)(b * 32 + u)) * (32 * 32 * 6);
    for (int i = t; i < 6144; i += 256) {
        int f = i % 6;
        int w = (i / 6) % 32;
        int v = i / 192;
        float r = 0.0f;
        #pragma unroll
        for (int kz = 0; kz < 5; ++kz) {
            int ti = (f * 5 + kz) * 32 + v;
            r += T2re[ti] * Fp[w * 5 + kz] + T2im[ti] * Fm[w * 5 + kz];
        }
        obase[i] = r;
    }
}

// ---------------------------------------------------------------------------
extern "C" void kernel_launch(void* const* d_in, const int* in_sizes, int n_in,
                              void* d_out, int out_size, void* d_ws, size_t ws_size,
                              hipStream_t stream) {
    const float* img  = (const float*)d_in[0];   // (32, 32768) f32
    const float* W    = (const float*)d_in[1];   // (32768, 750) f32
    const float* bias = (const float*)d_in[2];   // (750,) f32
    float* out  = (float*)d_out;                 // (32,32,32,32,6) f32
    float* xbuf = (float*)d_ws;                  // 32 x 752 f32 accumulator

    const int nacc = 32 * NP_PAD;
    zero_ws_kernel<<<(nacc + 255) / 256, 256, 0, stream>>>(xbuf, nacc);

    dim3 gg(NTILES, KCHUNKS);
    gemm_wmma_kernel<<<gg, 32, 0, stream>>>(img, W, xbuf);

    dim3 ge(32, 32);
    ifft_eval_kernel<<<ge, 256, 0, stream>>>(xbuf, bias, out);
}